// NeuroSATLoss_53730040873557
// MI455X (gfx1250) — compile-verified
//
#include <hip/hip_runtime.h>
#include <math.h>

// ---------------------------------------------------------------------------
// NeuroSAT loss: ragged segmented reduction (16384 segments, 33.5M f32 total)
// HBM-bound: 134 MB read @ 23.3 TB/s -> ~5.8us floor. One coalesced pass.
// ---------------------------------------------------------------------------

typedef float v2f __attribute__((ext_vector_type(2)));
typedef float v8f __attribute__((ext_vector_type(8)));

#define SCAN_THREADS 256
#define SEG_THREADS  256

// ---- Kernel 1: exclusive prefix sum of lit_sizes -> offsets (one block) ----
__global__ void nsat_scan_kernel(const int* __restrict__ sizes,
                                 int* __restrict__ offsets, int n) {
    __shared__ int lds[SCAN_THREADS];
    const int tid = threadIdx.x;
    const int per = (n + SCAN_THREADS - 1) / SCAN_THREADS;
    const int base = tid * per;
    const int end  = (base + per < n) ? (base + per) : n;

    int local = 0;
    for (int i = base; i < end; ++i) local += sizes[i];
    lds[tid] = local;
    __syncthreads();

    // Hillis-Steele inclusive scan over 256 partials (deterministic)
    for (int off = 1; off < SCAN_THREADS; off <<= 1) {
        int v = (tid >= off) ? lds[tid - off] : 0;
        __syncthreads();
        lds[tid] += v;
        __syncthreads();
    }

    int run = (tid == 0) ? 0 : lds[tid - 1];
    for (int i = base; i < end; ++i) {
        offsets[i] = run;
        run += sizes[i];
    }
}

// ---- Kernel 2: one block per segment; streaming b128 loads + dual sums -----
__global__ void nsat_seg_kernel(const float* __restrict__ pred,
                                const int* __restrict__ sizes,
                                const int* __restrict__ offsets,
                                const float* __restrict__ dw,
                                const float* __restrict__ db,
                                float* __restrict__ losses) {
    const int seg   = blockIdx.x;
    const int start = offsets[seg];
    const int size  = sizes[seg];
    const float* p  = pred + start;
    const int tid   = threadIdx.x;

    float s = 0.0f;  // sum(p)
    float q = 0.0f;  // sum((p-0.5)^2)

    // scalar head until 16B alignment
    int head = (4 - (start & 3)) & 3;
    if (head > size) head = size;
    for (int i = tid; i < head; i += SEG_THREADS) {
        float v = p[i];
        s += v;
        float d = v - 0.5f;
        q += d * d;
    }

    // vector body: float4 -> global_load_b128, fully coalesced
    const int nvec = (size - head) >> 2;
    const float4* __restrict__ pv = (const float4*)(p + head);
    for (int i = tid; i < nvec; i += SEG_THREADS) {
        float4 v = pv[i];
        s += v.x + v.y + v.z + v.w;
        float d0 = v.x - 0.5f, d1 = v.y - 0.5f;
        float d2 = v.z - 0.5f, d3 = v.w - 0.5f;
        q += d0 * d0 + d1 * d1 + d2 * d2 + d3 * d3;
    }

    // scalar tail
    for (int i = head + (nvec << 2) + tid; i < size; i += SEG_THREADS) {
        float v = p[i];
        s += v;
        float d = v - 0.5f;
        q += d * d;
    }

    // wave32 reduction (gfx1250 is wave32)
    for (int off = 16; off > 0; off >>= 1) {
        s += __shfl_down(s, off, 32);
        q += __shfl_down(q, off, 32);
    }

    __shared__ float ls[SEG_THREADS / 32];
    __shared__ float lq[SEG_THREADS / 32];
    const int wave = tid >> 5;
    const int lane = tid & 31;
    if (lane == 0) { ls[wave] = s; lq[wave] = q; }
    __syncthreads();

    if (tid == 0) {
        float S = 0.0f, Q = 0.0f;
        for (int i = 0; i < SEG_THREADS / 32; ++i) { S += ls[i]; Q += lq[i]; }
        const float fs   = (float)size;
        const float mean = S / fs;
        const float z    = dw[0] * mean + db[0];
        const float sig  = 1.0f / (1.0f + expf(-z));
        const float sgnl = 1.0f - 2.0f * sig;   // (sig*2-1)*-1
        losses[seg] = sgnl * Q / fs;
    }
}

// ---- Kernel 3: WMMA f32 reduction of 16384 losses (ones-matrix trick) -----
// A = 16x4 f32 tile of data (2 VGPRs/lane), B = all-ones 4x16 f32.
// D[m][n] = sum_k A[m][k] + C[m][n]; summing the full 16x16 accumulator
// gives 16x the grand total. Exact f32, fixed order, deterministic.
__global__ void nsat_final_reduce_wmma(const float* __restrict__ losses,
                                       float* __restrict__ out, int n) {
    const int lane = threadIdx.x;   // 32 lanes, one wave; EXEC all ones
    v8f c = {};
    v2f bones;
    bones.x = 1.0f;
    bones.y = 1.0f;

    const int nfull = n & ~63;      // 64 values consumed per WMMA
    for (int base = 0; base < nfull; base += 64) {
        float2 d = ((const float2*)(losses + base))[lane];
        v2f a;
        a.x = d.x;
        a.y = d.y;
        // 8 args: (neg_a, A, neg_b, B, c_mod, C, reuse_a, reuse_b)
        c = __builtin_amdgcn_wmma_f32_16x16x4_f32(
                false, a, false, bones, (short)0, c, false, false);
    }

    // remainder handled scalar (n=16384 -> empty, but stay generic)
    float extra = 0.0f;
    for (int i = nfull + lane; i < n; i += 32) extra += losses[i];

    float s = c[0] + c[1] + c[2] + c[3] + c[4] + c[5] + c[6] + c[7];
    for (int off = 16; off > 0; off >>= 1) {
        s     += __shfl_down(s, off, 32);
        extra += __shfl_down(extra, off, 32);
    }
    if (lane == 0) {
        out[0] = (s * (1.0f / 16.0f) + extra) / (float)n;
    }
}

// ---------------------------------------------------------------------------
extern "C" void kernel_launch(void* const* d_in, const int* in_sizes, int n_in,
                              void* d_out, int out_size, void* d_ws, size_t ws_size,
                              hipStream_t stream) {
    (void)n_in; (void)out_size; (void)ws_size;
    const float* pred  = (const float*)d_in[0];
    const int*   sizes = (const int*)d_in[1];
    const float* dw    = (const float*)d_in[2];
    const float* db    = (const float*)d_in[3];
    float*       out   = (float*)d_out;
    const int n = in_sizes[1];            // 16384 segments

    int*   offsets = (int*)d_ws;                                   // n * 4B
    float* losses  = (float*)((char*)d_ws + (size_t)n * sizeof(int)); // n * 4B

    nsat_scan_kernel<<<1, SCAN_THREADS, 0, stream>>>(sizes, offsets, n);
    nsat_seg_kernel<<<n, SEG_THREADS, 0, stream>>>(pred, sizes, offsets,
                                                   dw, db, losses);
    nsat_final_reduce_wmma<<<1, 32, 0, stream>>>(losses, out, n);
}